// AdaptiveTemporalConv_73409581023541
// MI455X (gfx1250) — compile-verified
//
#include <hip/hip_runtime.h>
#include <math.h>

#define T 1024
#define D 512
#define BSZ 8
#define WSPAN 16

typedef float v2f __attribute__((ext_vector_type(2)));
typedef float v8f __attribute__((ext_vector_type(8)));

// ---------------- Kernel 0: row L2 norms + min_span init ----------------
__global__ __launch_bounds__(256) void norms_kernel(const float* __restrict__ x,
                                                    float* __restrict__ norms,
                                                    int* __restrict__ minspan) {
  const int row = blockIdx.x;            // 0..8191  (b*T + i)
  const int tid = threadIdx.x;
  const float* xr = x + (size_t)row * D;
  float a0 = xr[tid];
  float a1 = xr[tid + 256];
  float s = a0 * a0 + a1 * a1;
  #pragma unroll
  for (int off = 16; off > 0; off >>= 1) s += __shfl_xor(s, off, 32);
  __shared__ float red[8];
  const int wave = tid >> 5, lane = tid & 31;
  if (lane == 0) red[wave] = s;
  __syncthreads();
  if (tid == 0) {
    float tot = 0.f;
    #pragma unroll
    for (int w = 0; w < 8; ++w) tot += red[w];
    norms[row] = sqrtf(tot);
    if (row == 0) *minspan = WSPAN - 1;   // max possible value of min(15, r_max)
  }
}

// ---------------- Kernel 1: banded cosine scores via WMMA -> r_max, min_span ----
// 4 waves per 16-row tile, split-K (128 each). Each wave prefetches its whole
// 512B K-chunk (global_prefetch_b8, no LOADcnt) so the MMA-loop loads are cache
// hits; partial 16x16 f32 accumulators reduced through LDS; wave 0 does softmax.
__global__ __launch_bounds__(128) void score_kernel(const float* __restrict__ x,
                                                    const float* __restrict__ norms,
                                                    int* __restrict__ rmaxbuf,
                                                    int* __restrict__ minspan) {
  const int tile = blockIdx.x;           // 0..511
  const int b    = tile >> 6;
  const int i0   = (tile & 63) << 4;
  const int tid  = threadIdx.x;
  const int wave = tid >> 5;             // 0..3  (K chunk)
  const int lane = tid & 31;
  const int half = lane >> 4;            // K sub-pair select
  const int n    = lane & 15;            // M for A, N for B
  const float* xb = x + (size_t)b * T * D;
  const float* nb = norms + (size_t)b * T;

  const int jbase = i0 - 16;             // j-window covers [i0-15, i0+31] within 48 cols
  const int j0 = jbase + n;
  const int j1 = j0 + 16;
  const int j2 = j0 + 32;
  const int j0c = min(max(j0, 0), T - 1);
  const int j1c = min(max(j1, 0), T - 1);
  const int j2c = min(max(j2, 0), T - 1);

  // A: lane n holds row i0+n, K = kbeg + 4*s + 2*half + {0,1}.  B mirrors, col n -> row j.
  const int kbeg = wave * (D / 4);       // 128-wide K chunk per wave
  const float* ap  = xb + (size_t)(i0 + n) * D + 2 * half + kbeg;
  const float* bp0 = xb + (size_t)j0c * D + 2 * half + kbeg;
  const float* bp1 = xb + (size_t)j1c * D + 2 * half + kbeg;
  const float* bp2 = xb + (size_t)j2c * D + 2 * half + kbeg;

  // Warm L0/L2 for this wave's whole K-chunk: 4 cachelines per pointer,
  // no register target, no LOADcnt tracking -> pure latency reduction.
  #pragma unroll
  for (int l = 0; l < 4; ++l) {
    __builtin_prefetch(ap  + 32 * l, 0, 3);
    __builtin_prefetch(bp0 + 32 * l, 0, 3);
    __builtin_prefetch(bp1 + 32 * l, 0, 3);
    __builtin_prefetch(bp2 + 32 * l, 0, 3);
  }

  v8f acc0 = {}; v8f acc1 = {}; v8f acc2 = {};
  v2f a  = *(const v2f*)(ap);
  v2f f0 = *(const v2f*)(bp0);
  v2f f1 = *(const v2f*)(bp1);
  v2f f2 = *(const v2f*)(bp2);
  #pragma unroll 4
  for (int s = 0; s < 31; ++s) {
    const int kn = 4 * (s + 1);
    v2f an = *(const v2f*)(ap  + kn);
    v2f g0 = *(const v2f*)(bp0 + kn);
    v2f g1 = *(const v2f*)(bp1 + kn);
    v2f g2 = *(const v2f*)(bp2 + kn);
    acc0 = __builtin_amdgcn_wmma_f32_16x16x4_f32(false, a, false, f0, (short)0, acc0, false, false);
    acc1 = __builtin_amdgcn_wmma_f32_16x16x4_f32(false, a, false, f1, (short)0, acc1, false, false);
    acc2 = __builtin_amdgcn_wmma_f32_16x16x4_f32(false, a, false, f2, (short)0, acc2, false, false);
    a = an; f0 = g0; f1 = g1; f2 = g2;
  }
  acc0 = __builtin_amdgcn_wmma_f32_16x16x4_f32(false, a, false, f0, (short)0, acc0, false, false);
  acc1 = __builtin_amdgcn_wmma_f32_16x16x4_f32(false, a, false, f1, (short)0, acc1, false, false);
  acc2 = __builtin_amdgcn_wmma_f32_16x16x4_f32(false, a, false, f2, (short)0, acc2, false, false);

  // Reduce split-K partials through LDS.
  __shared__ float part[4][3][8][32];
  #pragma unroll
  for (int r = 0; r < 8; ++r) {
    part[wave][0][r][lane] = acc0[r];
    part[wave][1][r][lane] = acc1[r];
    part[wave][2][r][lane] = acc2[r];
  }
  __syncthreads();
  if (wave != 0) return;
  #pragma unroll
  for (int r = 0; r < 8; ++r) {
    acc0[r] = (part[0][0][r][lane] + part[1][0][r][lane]) + (part[2][0][r][lane] + part[3][0][r][lane]);
    acc1[r] = (part[0][1][r][lane] + part[1][1][r][lane]) + (part[2][1][r][lane] + part[3][1][r][lane]);
    acc2[r] = (part[0][2][r][lane] + part[1][2][r][lane]) + (part[2][2][r][lane] + part[3][2][r][lane]);
  }

  // C layout: VGPR r = row (r + 8*half), column = lane&15 (+16*tile).
  // Each 16-lane half reduces its own row via xor-shuffles (masks 1..8 stay in-half).
  const int jj[3] = {j0, j1, j2};
  for (int r = 0; r < 8; ++r) {
    const int i  = i0 + r + 8 * half;
    const float ni = nb[i];
    float accs[3]; accs[0] = acc0[r]; accs[1] = acc1[r]; accs[2] = acc2[r];
    float sc[3]; bool ok[3]; int dd[3];
    float mx = -INFINITY;
    #pragma unroll
    for (int t = 0; t < 3; ++t) {
      const int j = jj[t];
      const int d = j - i;
      const bool v = (j >= 0) && (j < T) && (d >= -(WSPAN - 1)) && (d <= WSPAN) && (d != 0);
      const float nj = nb[min(max(j, 0), T - 1)];
      const float s = accs[t] / (ni * nj + 1e-8f);
      sc[t] = v ? s : -INFINITY;
      ok[t] = v; dd[t] = d;
      mx = fmaxf(mx, sc[t]);
    }
    #pragma unroll
    for (int off = 1; off < 16; off <<= 1) mx = fmaxf(mx, __shfl_xor(mx, off, 32));
    float sum = 0.f, rsum = 0.f;
    #pragma unroll
    for (int t = 0; t < 3; ++t) {
      const float e = ok[t] ? expf(sc[t] - mx) : 0.f;
      sum += e;
      if (dd[t] >= 1 && dd[t] <= WSPAN) rsum += e;
    }
    #pragma unroll
    for (int off = 1; off < 16; off <<= 1) {
      sum  += __shfl_xor(sum,  off, 32);
      rsum += __shfl_xor(rsum, off, 32);
    }
    const float rscore = rsum / sum;
    const int rspan = (int)(16.f * rscore);
    const int rmx = min(i + rspan, T);
    if (n == 0) {
      rmaxbuf[(size_t)b * T + i] = rmx;
      atomicMin(minspan, min(rmx, WSPAN - 1));
    }
  }
}

// ---------------- Kernel 2: gather + rel_emb + (dwconv3 -> LN_c -> relu) x2 -> max ----
__device__ inline void ln_relu(float* h, int ns, float g, float be,
                               float (*red)[13][2], float (*stat)[2], int c) {
  const int wave = c >> 5, lane = c & 31;
  __syncthreads();                       // protect LDS reuse across calls
  for (int s = 0; s < ns; ++s) {
    float v = h[s], v2 = v * v;
    #pragma unroll
    for (int off = 16; off > 0; off >>= 1) {
      v  += __shfl_xor(v,  off, 32);
      v2 += __shfl_xor(v2, off, 32);
    }
    if (lane == 0) { red[wave][s][0] = v; red[wave][s][1] = v2; }
  }
  __syncthreads();
  if (c < ns) {
    float s1 = 0.f, s2 = 0.f;
    #pragma unroll
    for (int w = 0; w < 16; ++w) { s1 += red[w][c][0]; s2 += red[w][c][1]; }
    const float mean = s1 * (1.f / D);
    const float var  = s2 * (1.f / D) - mean * mean;
    stat[c][0] = mean;
    stat[c][1] = 1.f / sqrtf(var + 1e-5f);
  }
  __syncthreads();
  for (int s = 0; s < ns; ++s) {
    const float v = (h[s] - stat[s][0]) * stat[s][1] * g + be;
    h[s] = fmaxf(v, 0.f);
  }
}

__global__ __launch_bounds__(512) void conv_kernel(
    const float* __restrict__ x, const float* __restrict__ rel_emb,
    const float* __restrict__ c1w, const float* __restrict__ c1b,
    const float* __restrict__ g1,  const float* __restrict__ be1,
    const float* __restrict__ c2w, const float* __restrict__ c2b,
    const float* __restrict__ g2,  const float* __restrict__ be2,
    const int* __restrict__ rmaxbuf, const int* __restrict__ minspan,
    float* __restrict__ out) {
  const int bi = blockIdx.x;             // 0..8191
  const int b  = bi >> 10;
  const int i  = bi & (T - 1);
  const int c  = threadIdx.x;            // channel
  const int k  = *minspan;               // global k, <= 15
  const int rm  = rmaxbuf[bi];
  const int jhi = min(rm, T - 1);
  const int jlo = jhi - k + 1;           // ids = jlo..jhi (contiguous; top_k+sort collapses)
  const float* xb = x + (size_t)b * T * D;

  float nf[15];
  #pragma unroll
  for (int s = 0; s < 15; ++s) {
    if (s < k) {
      const int j = jlo + s;
      const int rel = min(max(j - i + WSPAN, 0), 2 * WSPAN - 1);
      nf[s] = xb[(size_t)j * D + c] + rel_emb[(size_t)rel * D + c];
    } else nf[s] = 0.f;
  }

  __shared__ float red[16][13][2];
  __shared__ float stat[13][2];

  // depthwise conv1 (VALID, width 3)
  const float a0 = c1w[3 * c], a1 = c1w[3 * c + 1], a2 = c1w[3 * c + 2];
  const float cb1 = c1b[c];
  const int n1 = k - 2;
  float h1[13];
  #pragma unroll
  for (int s = 0; s < 13; ++s)
    h1[s] = (s < n1) ? fmaf(nf[s], a0, fmaf(nf[s + 1], a1, fmaf(nf[s + 2], a2, cb1))) : 0.f;
  ln_relu(h1, n1, g1[c], be1[c], red, stat, c);

  // depthwise conv2
  const float d0 = c2w[3 * c], d1 = c2w[3 * c + 1], d2 = c2w[3 * c + 2];
  const float cb2 = c2b[c];
  const int n2 = n1 - 2;
  float h2[11];
  #pragma unroll
  for (int s = 0; s < 11; ++s)
    h2[s] = (s < n2) ? fmaf(h1[s], d0, fmaf(h1[s + 1], d1, fmaf(h1[s + 2], d2, cb2))) : 0.f;
  ln_relu(h2, n2, g2[c], be2[c], red, stat, c);

  float m = -INFINITY;
  #pragma unroll
  for (int s = 0; s < 11; ++s)
    if (s < n2) m = fmaxf(m, h2[s]);
  if (n2 <= 0) m = 0.f;
  out[(size_t)bi * D + c] = m;
}

extern "C" void kernel_launch(void* const* d_in, const int* in_sizes, int n_in,
                              void* d_out, int out_size, void* d_ws, size_t ws_size,
                              hipStream_t stream) {
  const float* x       = (const float*)d_in[0];
  const float* rel_emb = (const float*)d_in[1];
  const float* c1w     = (const float*)d_in[2];
  const float* c1b     = (const float*)d_in[3];
  const float* g1      = (const float*)d_in[4];
  const float* be1     = (const float*)d_in[5];
  const float* c2w     = (const float*)d_in[6];
  const float* c2b     = (const float*)d_in[7];
  const float* g2      = (const float*)d_in[8];
  const float* be2     = (const float*)d_in[9];
  float* out = (float*)d_out;

  float* norms  = (float*)d_ws;                                      // 8192 f32
  int*   rmaxb  = (int*)((char*)d_ws + (size_t)BSZ * T * 4);         // 8192 i32
  int*   minspn = (int*)((char*)d_ws + (size_t)2 * BSZ * T * 4);     // 1 i32

  norms_kernel<<<BSZ * T, 256, 0, stream>>>(x, norms, minspn);
  score_kernel<<<BSZ * (T / 16), 128, 0, stream>>>(x, norms, rmaxb, minspn);
  conv_kernel<<<BSZ * T, 512, 0, stream>>>(x, rel_emb, c1w, c1b, g1, be1,
                                           c2w, c2b, g2, be2, rmaxb, minspn, out);
}